// CRF_6597069767075
// MI455X (gfx1250) — compile-verified
//
#include <hip/hip_runtime.h>
#include <hip/hip_bf16.h>

typedef __attribute__((ext_vector_type(2))) float v2f;
typedef __attribute__((ext_vector_type(8))) float v8f;

#define CRF_B 32
#define CRF_T 512
#define CRF_K 256
#define PADK  258   // pad LDS rows (258 % 64 == 2) to dodge bank conflicts

// ---------------------------------------------------------------------------
// E[i,j] = exp(trans[i,j])  (time-invariant linear-space transition matrix)
// ---------------------------------------------------------------------------
__global__ void crf_exp_trans(const float* __restrict__ trans, float* __restrict__ E) {
    int i = blockIdx.x * blockDim.x + threadIdx.x;
    if (i < CRF_K * CRF_K) E[i] = __expf(trans[i]);
}

// ---------------------------------------------------------------------------
// Joint (numerator) score per batch: sum_t emit[b,t,tag] * mask
//                                  + sum_{t>=1} trans[tag_{t-1}, tag_t] * mask
// ---------------------------------------------------------------------------
__global__ void crf_numerator(const float* __restrict__ emissions,
                              const int* __restrict__ tags,
                              const unsigned char* __restrict__ mask,
                              const float* __restrict__ trans,
                              float* __restrict__ log_num) {
    const int b = blockIdx.x;
    float acc = 0.f;
    for (int t = threadIdx.x; t < CRF_T; t += blockDim.x) {
        const int tag = tags[b * CRF_T + t];
        const float mk = mask[b * CRF_T + t] ? 1.f : 0.f;
        acc += emissions[((size_t)b * CRF_T + t) * CRF_K + tag] * mk;
        if (t >= 1) {
            const int pt = tags[b * CRF_T + t - 1];
            acc += trans[pt * CRF_K + tag] * mk;
        }
    }
#pragma unroll
    for (int off = 16; off; off >>= 1) acc += __shfl_xor(acc, off, 32);
    __shared__ float red[16];
    const int lane = threadIdx.x & 31, wave = threadIdx.x >> 5;
    if (lane == 0) red[wave] = acc;
    __syncthreads();
    if (threadIdx.x == 0) {
        float s = 0.f;
        for (int w = 0; w < (int)(blockDim.x >> 5); ++w) s += red[w];
        log_num[b] = s;
    }
}

// ---------------------------------------------------------------------------
// Forward recursion (denominator). One workgroup per 16 batches; 16 waves.
// Wave w owns output columns [16w, 16w+16) -> one 16x16 f32 WMMA C-tile.
// B-operand slice of E (256x16 f32 = 64 float2 fragments) is register-resident.
// Per step: v = exp(alpha - max) in LDS, C = v @ E via 64x V_WMMA_F32_16X16X4_F32,
// alpha' = log(C) + max + emit_t (masked).
// ---------------------------------------------------------------------------
__global__ __launch_bounds__(512)
void crf_forward(const float* __restrict__ emissions,
                 const unsigned char* __restrict__ mask,
                 const float* __restrict__ E,
                 float* __restrict__ log_den) {
    __shared__ float alpha[16][PADK];
    __shared__ float vlds[16][PADK];
    __shared__ float mlds[16];

    const int tid  = threadIdx.x;
    const int lane = tid & 31;
    const int wave = tid >> 5;           // 0..15: owns batch `wave` for reductions
    const int mb0  = blockIdx.x * 16;    // first batch handled by this workgroup
    const int half = lane >> 4;          // 0 or 1 (lane half)
    const int kk2  = half * 2;           // K sub-offset inside a 4-wide fragment
    const int nn   = lane & 15;          // N column within tile / M row for A-frag

    // ---- one-time: load this wave's B-slice of E into registers (128 VGPRs) ----
    const int j = wave * 16 + nn;        // global "to"-state column owned by lane
    v2f bfrag[64];
#pragma unroll
    for (int kk = 0; kk < 64; ++kk) {
        const int k0 = kk * 4 + kk2;
        v2f b;
        b.x = E[(size_t)(k0 + 0) * CRF_K + j];
        b.y = E[(size_t)(k0 + 1) * CRF_K + j];
        bfrag[kk] = b;
    }

    // ---- alpha_0 = emissions[:, 0, :] ----
    {
        const float* em0 = emissions + (size_t)(mb0 + wave) * CRF_T * CRF_K;
        for (int k = lane; k < CRF_K; k += 32) alpha[wave][k] = em0[k];
    }
    __syncthreads();

    for (int t = 1; t < CRF_T; ++t) {
        // ---- wave `b` computes max_b and v = exp(alpha_b - max_b) ----
        const int b = wave;
        float mx = -3.0e38f;
        for (int k = lane; k < CRF_K; k += 32) mx = fmaxf(mx, alpha[b][k]);
#pragma unroll
        for (int off = 16; off; off >>= 1) mx = fmaxf(mx, __shfl_xor(mx, off, 32));
        if (lane == 0) mlds[b] = mx;
        for (int k = lane; k < CRF_K; k += 32) vlds[b][k] = __expf(alpha[b][k] - mx);

        // prefetch next step's emission rows (lane<16: batch row, wave: j-segment)
        if (lane < 16 && (t + 1) < CRF_T) {
            const float* p = emissions +
                ((size_t)(mb0 + lane) * CRF_T + (t + 1)) * CRF_K + wave * 16;
            __builtin_prefetch(p, 0, 3);
        }
        __syncthreads();

        // ---- C(16x16) = v(16x256) @ E_slice(256x16), f32 WMMA K-loop ----
        v8f c = {};
#pragma unroll
        for (int kk = 0; kk < 64; ++kk) {
            const int k0 = kk * 4 + kk2;
            v2f a;
            a.x = vlds[nn][k0];
            a.y = vlds[nn][k0 + 1];
            c = __builtin_amdgcn_wmma_f32_16x16x4_f32(false, a, false, bfrag[kk],
                                                      (short)0, c, false, false);
        }

        // ---- alpha_new[m][j] = log(C[m][j]) + max_m + emit, where masked ----
#pragma unroll
        for (int r = 0; r < 8; ++r) {
            const int m = r + 8 * half;  // batch row of this C element
            const unsigned char mk = mask[(size_t)(mb0 + m) * CRF_T + t];
            if (mk) {
                const float emit =
                    emissions[((size_t)(mb0 + m) * CRF_T + t) * CRF_K + wave * 16 + nn];
                alpha[m][wave * 16 + nn] = __logf(c[r]) + mlds[m] + emit;
            }
        }
        __syncthreads();
    }

    // ---- log_den[b] = logsumexp_j alpha[b][j] ----
    {
        const int b = wave;
        float mx = -3.0e38f;
        for (int k = lane; k < CRF_K; k += 32) mx = fmaxf(mx, alpha[b][k]);
#pragma unroll
        for (int off = 16; off; off >>= 1) mx = fmaxf(mx, __shfl_xor(mx, off, 32));
        float s = 0.f;
        for (int k = lane; k < CRF_K; k += 32) s += __expf(alpha[b][k] - mx);
#pragma unroll
        for (int off = 16; off; off >>= 1) s += __shfl_xor(s, off, 32);
        if (lane == 0) log_den[mb0 + b] = mx + __logf(s);
    }
}

// ---------------------------------------------------------------------------
// out = mean(log_den - log_num)
// ---------------------------------------------------------------------------
__global__ void crf_finalize(const float* __restrict__ log_den,
                             const float* __restrict__ log_num,
                             float* __restrict__ out) {
    float d = log_den[threadIdx.x] - log_num[threadIdx.x];
#pragma unroll
    for (int off = 16; off; off >>= 1) d += __shfl_xor(d, off, 32);
    if (threadIdx.x == 0) out[0] = d * (1.0f / CRF_B);
}

extern "C" void kernel_launch(void* const* d_in, const int* in_sizes, int n_in,
                              void* d_out, int out_size, void* d_ws, size_t ws_size,
                              hipStream_t stream) {
    const float*         emissions = (const float*)d_in[0];
    const int*           tags      = (const int*)d_in[1];
    const unsigned char* mask      = (const unsigned char*)d_in[2];
    const float*         trans     = (const float*)d_in[3];

    float* ws      = (float*)d_ws;
    float* E       = ws;                        // 256*256 f32 = 256 KB
    float* log_num = ws + CRF_K * CRF_K;        // 32 f32
    float* log_den = log_num + CRF_B;           // 32 f32
    float* out     = (float*)d_out;

    crf_exp_trans<<<(CRF_K * CRF_K + 255) / 256, 256, 0, stream>>>(trans, E);
    crf_numerator<<<CRF_B, 256, 0, stream>>>(emissions, tags, mask, trans, log_num);
    crf_forward<<<CRF_B / 16, 512, 0, stream>>>(emissions, mask, E, log_den);
    crf_finalize<<<1, 32, 0, stream>>>(log_den, log_num, out);
}